// EdgeConvOp_61976378081386
// MI455X (gfx1250) — compile-verified
//
#include <hip/hip_runtime.h>
#include <hip/hip_bf16.h>

#define NNODES 50000
#define NEDGES 600000
#define FEAT   128
#define NLAYERS 4

typedef float v2f __attribute__((ext_vector_type(2)));
typedef float v8f __attribute__((ext_vector_type(8)));

// ---------------------------------------------------------------------------
// Y[n][o] = sum_k X[n][k] * W[o][k]  (+ bias[o] if bias != nullptr)
// One wave computes one 16x16 tile of Y via 32 x V_WMMA_F32_16X16X4_F32.
// blockDim = 256 (8 waves); wave w -> column tile w (8*16 = 128 = FEAT cols),
// blockIdx.x -> row tile (NNODES/16 = 3125 blocks, exact).
// ---------------------------------------------------------------------------
__global__ __launch_bounds__(256) void gemm_nt_wmma(
    const float* __restrict__ X, const float* __restrict__ W,
    const float* __restrict__ bias, float* __restrict__ Y) {
  const int lane = threadIdx.x & 31;
  const int wave = threadIdx.x >> 5;
  const int half = lane >> 4;     // 0: lanes 0-15, 1: lanes 16-31
  const int r    = lane & 15;
  const long row0 = (long)blockIdx.x * 16;
  const int  col0 = wave * 16;

  const float* arow = X + (row0 + r) * FEAT;          // A: row row0+r
  const float* brow = W + (long)(col0 + r) * FEAT;    // B(k,o)=W[o][k]: row col0+r of W

  v8f c = {};
  #pragma unroll
  for (int k0 = 0; k0 < FEAT; k0 += 4) {
    const int kk = k0 + 2 * half;                     // K split: v0->K0/K2, v1->K1/K3
    const v2f a = *(const v2f*)(arow + kk);
    const v2f b = *(const v2f*)(brow + kk);
    c = __builtin_amdgcn_wmma_f32_16x16x4_f32(false, a, false, b, (short)0, c,
                                              false, false);
  }

  const float bv = (bias != nullptr) ? bias[col0 + r] : 0.0f;
  #pragma unroll
  for (int v = 0; v < 8; ++v) {
    // C/D layout: VGPR v -> (M = v + 8*half, N = r)
    Y[(row0 + v + 8 * half) * FEAT + col0 + r] = c[v] + bv;
  }
}

// ---------------------------------------------------------------------------
// Float atomic max via monotone bit-pattern trick:
//   non-negative floats: int-compare order == float order  -> atomicMax(int)
//   negative floats: uint order is reversed float order    -> atomicMin(uint)
// Init value -inf (0xFF800000) is the largest finite-neg uint pattern.
// ---------------------------------------------------------------------------
__device__ __forceinline__ void atomic_max_f32(float* addr, float val) {
  const int bits = __float_as_int(val);
  if (bits >= 0) {
    atomicMax((int*)addr, bits);
  } else {
    atomicMin((unsigned int*)addr, (unsigned int)bits);
  }
}

// One wave per edge: 32 lanes x float4 = 128 features.
__global__ __launch_bounds__(256) void scatter_max_kernel(
    const float* __restrict__ T, const int* __restrict__ src,
    const int* __restrict__ dst, float* __restrict__ M) {
  const int wid  = (blockIdx.x * 256 + (int)threadIdx.x) >> 5;
  const int lane = threadIdx.x & 31;
  if (wid >= NEDGES) return;
  const int s = src[wid];
  const int d = dst[wid];
  const float4 v = *(const float4*)(T + (long)s * FEAT + lane * 4);
  float* mp = M + (long)d * FEAT + lane * 4;
  atomic_max_f32(mp + 0, v.x);
  atomic_max_f32(mp + 1, v.y);
  atomic_max_f32(mp + 2, v.z);
  atomic_max_f32(mp + 3, v.w);
}

__global__ __launch_bounds__(256) void init_neg_inf(float4* __restrict__ M) {
  const long i = (long)blockIdx.x * 256 + threadIdx.x;   // exactly N*F/4 threads
  const float ninf = -__builtin_inff();
  M[i] = make_float4(ninf, ninf, ninf, ninf);
}

// out = (m == -inf) ? 0 : relu(m - t + tb + phi)
__global__ __launch_bounds__(256) void combine_kernel(
    const float* __restrict__ M, const float* __restrict__ T,
    const float* __restrict__ PHI, const float* __restrict__ tb,
    float* __restrict__ Y) {
  const long i4   = (long)blockIdx.x * 256 + threadIdx.x; // float4 index, exact
  const long base = i4 * 4;
  const int  f    = (int)(base & (FEAT - 1));
  const float4 mv = *(const float4*)(M + base);
  const float4 tv = *(const float4*)(T + base);
  const float4 pv = *(const float4*)(PHI + base);
  const float ninf = -__builtin_inff();
  float4 o;
  o.x = (mv.x == ninf) ? 0.0f : fmaxf(0.0f, mv.x - tv.x + tb[f + 0] + pv.x);
  o.y = (mv.y == ninf) ? 0.0f : fmaxf(0.0f, mv.y - tv.y + tb[f + 1] + pv.y);
  o.z = (mv.z == ninf) ? 0.0f : fmaxf(0.0f, mv.z - tv.z + tb[f + 2] + pv.z);
  o.w = (mv.w == ninf) ? 0.0f : fmaxf(0.0f, mv.w - tv.w + tb[f + 3] + pv.w);
  *(float4*)(Y + base) = o;
}

extern "C" void kernel_launch(void* const* d_in, const int* in_sizes, int n_in,
                              void* d_out, int out_size, void* d_ws, size_t ws_size,
                              hipStream_t stream) {
  (void)in_sizes; (void)n_in; (void)out_size; (void)ws_size;

  const float* feats   = (const float*)d_in[0];
  const int*   src     = (const int*)d_in[1];
  const int*   dst     = (const int*)d_in[2];
  const float* theta_w = (const float*)d_in[3];
  const float* theta_b = (const float*)d_in[4];
  const float* phi_w   = (const float*)d_in[5];
  const float* phi_b   = (const float*)d_in[6];
  float* out = (float*)d_out;

  const size_t be = (size_t)NNODES * FEAT;     // 6.4M floats = 25.6 MB per buffer
  float* T   = (float*)d_ws;                   // t   = x @ tw.T
  float* PHI = T + be;                         // phi = x @ pw.T + pb
  float* M   = PHI + be;                       // segment max of t[src] over dst
  float* XB  = M + be;                         // ping buffer for layer outputs

  const int gemm_grid  = NNODES / 16;          // 3125
  const int elem_grid  = (NNODES * FEAT / 4) / 256; // 6250
  const int edge_grid  = NEDGES / 8;           // 75000 (8 waves/block, 1 wave/edge)

  const float* x = feats;
  for (int l = 0; l < NLAYERS; ++l) {
    float* y = (l == NLAYERS - 1) ? out : XB;
    const float* tw = theta_w + (size_t)l * FEAT * FEAT;
    const float* pw = phi_w   + (size_t)l * FEAT * FEAT;

    gemm_nt_wmma<<<gemm_grid, 256, 0, stream>>>(x, tw, nullptr, T);
    gemm_nt_wmma<<<gemm_grid, 256, 0, stream>>>(x, pw, phi_b + (size_t)l * FEAT, PHI);
    init_neg_inf<<<elem_grid, 256, 0, stream>>>((float4*)M);
    scatter_max_kernel<<<edge_grid, 256, 0, stream>>>(T, src, dst, M);
    combine_kernel<<<elem_grid, 256, 0, stream>>>(M, T, PHI,
                                                  theta_b + (size_t)l * FEAT, y);
    x = y;
  }
}